// GAT_5342939316734
// MI455X (gfx1250) — compile-verified
//
#include <hip/hip_runtime.h>
#include <math.h>

// ---------------------------------------------------------------------------
// GAT (2-layer, PyG-style GATConv, eval mode) for MI455X / gfx1250.
//   N = 100000 nodes, E = 1.6M edges (+N self loops), IN=128, HID=128 (8x16),
//   OUT=64 (1x64).
// GEMMs: V_WMMA_F32_16X16X4_F32, W staged in LDS, one wave per 16-row stripe
//        computing all N/16 column tiles (A fragment amortized over NT WMMAs).
// Edge phase: unrolled gathers (b128 clauses) + fp32 global atomics.
// ---------------------------------------------------------------------------

typedef __attribute__((ext_vector_type(2))) float v2f;
typedef __attribute__((ext_vector_type(8))) float v8f;

// ---------------------------------------------------------------------------
// Dense GEMM  C[M,N] = A[M,K] * B[K,N],  K=128 fixed, N = NT*16.
// Block = 256 threads (8 waves). B is loaded into LDS once per block; each
// wave then owns rows [tm,tm+16) and accumulates NT 16x16 WMMA tiles.
// Fragment layouts per CDNA5 ISA 7.12.2 (32-bit A 16x4 / B 4x16 / C 16x16).
// ---------------------------------------------------------------------------
template <int K, int NT>
__global__ void gat_gemm_wmma_f32(const float* __restrict__ A,
                                  const float* __restrict__ B,
                                  float* __restrict__ C, int M) {
  constexpr int N = NT * 16;
  __shared__ float Bs[K * N];  // 64KB (NT=8) / 32KB (NT=4)

  // Cooperative load of the full B matrix (all threads participate, then
  // barrier, THEN out-of-range waves may exit -- keeps barrier converged).
  {
    const float4* __restrict__ B4 = (const float4*)B;
    float4* Bs4 = (float4*)Bs;
    const int total4 = (K * N) >> 2;
    for (int i = (int)threadIdx.x; i < total4; i += (int)blockDim.x)
      Bs4[i] = B4[i];
  }
  __syncthreads();

  const int wave = (int)((blockIdx.x * blockDim.x + threadIdx.x) >> 5);
  const int lane = (int)(threadIdx.x & 31);
  const int rowTiles = M >> 4;
  if (wave >= rowTiles) return;  // wave-uniform: EXEC all-ones below

  const int tm = wave << 4;
  const int half = lane >> 4;  // 0: K+0/K+1,  1: K+2/K+3
  const int l15 = lane & 15;   // A: row in tile, B: col in tile
  const int kb = half << 1;

  const float* __restrict__ arow = A + (size_t)(tm + l15) * K;

  v8f acc[NT];
#pragma unroll
  for (int t = 0; t < NT; ++t) acc[t] = (v8f){};

  for (int k = 0; k < K; k += 4) {
    v2f a;
    a.x = arow[k + kb];      // merges into one global_load_b64
    a.y = arow[k + kb + 1];
    const float* __restrict__ b0 = Bs + (k + kb) * N + l15;
#pragma unroll
    for (int t = 0; t < NT; ++t) {
      v2f b;
      b.x = b0[t * 16];      // ds_load_b32
      b.y = b0[N + t * 16];  // ds_load_b32 (next K row)
      acc[t] = __builtin_amdgcn_wmma_f32_16x16x4_f32(
          /*neg_a=*/false, a, /*neg_b=*/false, b,
          /*c_mod=*/(short)0, acc[t], /*reuse_a=*/false, /*reuse_b=*/false);
    }
  }

  // C/D layout: VGPR v -> row (half*8 + v), lanes 0..15 -> col within tile.
#pragma unroll
  for (int t = 0; t < NT; ++t) {
#pragma unroll
    for (int v = 0; v < 8; ++v) {
      C[(size_t)(tm + half * 8 + v) * N + t * 16 + l15] = acc[t][v];
    }
  }
}

// ---------------------------------------------------------------------------
// a_s[n,h] = dot(h[n,h,:], att_src[h,:]);  a_d likewise.  C templated so the
// dot unrolls into float4 (b128) loads.
// ---------------------------------------------------------------------------
template <int C>
__global__ void gat_attn_scores(const float* __restrict__ Hm,
                                const float* __restrict__ att_s,
                                const float* __restrict__ att_d,
                                float* __restrict__ a_s,
                                float* __restrict__ a_d, int Nn, int Hh) {
  const int i = (int)(blockIdx.x * blockDim.x + threadIdx.x);
  if (i >= Nn * Hh) return;
  const int h = i % Hh;
  const float4* __restrict__ row = (const float4*)(Hm + (size_t)i * C);
  const float4* __restrict__ as = (const float4*)(att_s + (size_t)h * C);
  const float4* __restrict__ ad = (const float4*)(att_d + (size_t)h * C);
  float ss = 0.f, sd = 0.f;
#pragma unroll
  for (int c = 0; c < C / 4; ++c) {
    const float4 v = row[c], s4 = as[c], d4 = ad[c];
    ss += v.x * s4.x + v.y * s4.y + v.z * s4.z + v.w * s4.w;
    sd += v.x * d4.x + v.y * d4.y + v.z * d4.z + v.w * d4.w;
  }
  a_s[i] = ss;
  a_d[i] = sd;
}

// ---------------------------------------------------------------------------
// Init helpers.
// ---------------------------------------------------------------------------
__global__ void gat_init_stats(float* __restrict__ mx, float* __restrict__ den,
                               int n) {
  const int i = (int)(blockIdx.x * blockDim.x + threadIdx.x);
  if (i < n) {
    mx[i] = -INFINITY;
    den[i] = 0.f;
  }
}

__global__ void gat_init_bias(float* __restrict__ out,
                              const float* __restrict__ bias, int Nn, int F) {
  const int i = (int)(blockIdx.x * blockDim.x + threadIdx.x);
  if (i < Nn * F) out[i] = bias[i % F];
}

// ---------------------------------------------------------------------------
// Ordered float atomic max (IEEE order preserved by int/uint trick).
// ---------------------------------------------------------------------------
__device__ __forceinline__ void atomicMaxF(float* addr, float val) {
  if (val >= 0.f)
    atomicMax((int*)addr, __float_as_int(val));
  else
    atomicMin((unsigned int*)addr, __float_as_uint(val));
}

// ---------------------------------------------------------------------------
// Edge pass 1: leaky-relu logits + segment max over dst.
// Self loops: e in [E, ET) -> src = dst = e - E.
// ---------------------------------------------------------------------------
template <int Hh>
__global__ void gat_edge_logits(const int* __restrict__ src,
                                const int* __restrict__ dst, int E, int ET,
                                const float* __restrict__ a_s,
                                const float* __restrict__ a_d,
                                float* __restrict__ logits,
                                float* __restrict__ mx) {
  const int e = (int)(blockIdx.x * blockDim.x + threadIdx.x);
  if (e >= ET) return;
  const int s = (e < E) ? src[e] : (e - E);
  const int d = (e < E) ? dst[e] : (e - E);
#pragma unroll
  for (int h = 0; h < Hh; ++h) {
    float l = a_s[(size_t)s * Hh + h] + a_d[(size_t)d * Hh + h];
    l = (l > 0.f) ? l : 0.2f * l;  // leaky_relu, slope 0.2
    logits[(size_t)e * Hh + h] = l;
    atomicMaxF(&mx[(size_t)d * Hh + h], l);
  }
}

// ---------------------------------------------------------------------------
// Edge pass 2: e = exp(logit - mx[dst]);  denom[dst] += e (stores e in place).
// ---------------------------------------------------------------------------
template <int Hh>
__global__ void gat_edge_exp(const int* __restrict__ dst, int E, int ET,
                             float* __restrict__ logits,
                             const float* __restrict__ mx,
                             float* __restrict__ den) {
  const int e = (int)(blockIdx.x * blockDim.x + threadIdx.x);
  if (e >= ET) return;
  const int d = (e < E) ? dst[e] : (e - E);
#pragma unroll
  for (int h = 0; h < Hh; ++h) {
    const float v = __expf(logits[(size_t)e * Hh + h] - mx[(size_t)d * Hh + h]);
    logits[(size_t)e * Hh + h] = v;
    atomicAdd(&den[(size_t)d * Hh + h], v);
  }
}

// ---------------------------------------------------------------------------
// Edge pass 3: out[dst,h,:] += h[src,h,:] * (e / denom[dst,h]).
// One thread per (edge, head); gathers C channels via float4 (b128) loads.
// ---------------------------------------------------------------------------
template <int Hh, int C>
__global__ void gat_edge_agg(const int* __restrict__ src,
                             const int* __restrict__ dst, int E, int ET,
                             const float* __restrict__ Hm,
                             const float* __restrict__ logits,
                             const float* __restrict__ den,
                             float* __restrict__ out) {
  const long long i = (long long)blockIdx.x * blockDim.x + threadIdx.x;
  if (i >= (long long)ET * Hh) return;
  const int e = (int)(i / Hh);
  const int h = (int)(i % Hh);
  const int s = (e < E) ? src[e] : (e - E);
  const int d = (e < E) ? dst[e] : (e - E);
  const float alpha = logits[i] / den[(size_t)d * Hh + h];
  const float4* __restrict__ hr =
      (const float4*)(Hm + ((size_t)s * Hh + h) * C);
  float* __restrict__ orow = out + ((size_t)d * Hh + h) * C;
#pragma unroll
  for (int c = 0; c < C / 4; ++c) {
    const float4 v = hr[c];
    atomicAdd(&orow[4 * c + 0], v.x * alpha);
    atomicAdd(&orow[4 * c + 1], v.y * alpha);
    atomicAdd(&orow[4 * c + 2], v.z * alpha);
    atomicAdd(&orow[4 * c + 3], v.w * alpha);
  }
}

// ---------------------------------------------------------------------------
// ELU in place (layer-1 activation).
// ---------------------------------------------------------------------------
__global__ void gat_elu(float* __restrict__ x, int n) {
  const int i = (int)(blockIdx.x * blockDim.x + threadIdx.x);
  if (i < n) {
    const float v = x[i];
    x[i] = (v > 0.f) ? v : (__expf(v) - 1.f);
  }
}

// ---------------------------------------------------------------------------
// Launcher.
// ---------------------------------------------------------------------------
extern "C" void kernel_launch(void* const* d_in, const int* in_sizes, int n_in,
                              void* d_out, int out_size, void* d_ws,
                              size_t ws_size, hipStream_t stream) {
  const float* x        = (const float*)d_in[0];   // [N,128]
  const int*   eidx     = (const int*)  d_in[1];   // [2,E]
  const float* W1       = (const float*)d_in[2];   // [128,128]
  const float* att_src1 = (const float*)d_in[3];   // [8,16]
  const float* att_dst1 = (const float*)d_in[4];   // [8,16]
  const float* b1       = (const float*)d_in[5];   // [128]
  const float* W2       = (const float*)d_in[6];   // [128,64]
  const float* att_src2 = (const float*)d_in[7];   // [1,64]
  const float* att_dst2 = (const float*)d_in[8];   // [1,64]
  const float* b2       = (const float*)d_in[9];   // [64]

  const int IN = 128, HID = 128, OUT = 64, H1 = 8;
  const int N = in_sizes[0] / IN;
  const int E = in_sizes[1] / 2;
  const int ET = E + N;  // with self loops

  const int* src = eidx;
  const int* dst = eidx + E;

  // Workspace carve-up (floats). Layer 2 reuses layer-1 regions.
  float* ws = (float*)d_ws;
  size_t off = 0;
  float* h1     = ws + off; off += (size_t)N * HID;   // also h2 [N,64]
  float* out1   = ws + off; off += (size_t)N * HID;   // layer-2 input
  float* a_s    = ws + off; off += (size_t)N * H1;
  float* a_d    = ws + off; off += (size_t)N * H1;
  float* mx     = ws + off; off += (size_t)N * H1;
  float* den    = ws + off; off += (size_t)N * H1;
  float* logits = ws + off; off += (size_t)ET * H1;
  (void)ws_size; (void)n_in; (void)out_size;

  const int TB = 256;  // 8 waves per block
  auto cdiv = [](long long a, long long b) { return (int)((a + b - 1) / b); };

  // ---------------- Layer 1 (H=8, C=16) ----------------
  {
    const int rowTiles = N >> 4;  // waves needed (one 16-row stripe each)
    gat_gemm_wmma_f32<128, 8><<<cdiv(rowTiles, 8), TB, 0, stream>>>(
        x, W1, h1, N);
    gat_attn_scores<16><<<cdiv((long long)N * H1, TB), TB, 0, stream>>>(
        h1, att_src1, att_dst1, a_s, a_d, N, H1);
    gat_init_stats<<<cdiv((long long)N * H1, TB), TB, 0, stream>>>(mx, den,
                                                                   N * H1);
    gat_init_bias<<<cdiv((long long)N * HID, TB), TB, 0, stream>>>(out1, b1, N,
                                                                   HID);
    gat_edge_logits<8><<<cdiv(ET, TB), TB, 0, stream>>>(src, dst, E, ET, a_s,
                                                        a_d, logits, mx);
    gat_edge_exp<8><<<cdiv(ET, TB), TB, 0, stream>>>(dst, E, ET, logits, mx,
                                                     den);
    gat_edge_agg<8, 16><<<cdiv((long long)ET * H1, TB), TB, 0, stream>>>(
        src, dst, E, ET, h1, logits, den, out1);
    gat_elu<<<cdiv((long long)N * HID, TB), TB, 0, stream>>>(out1, N * HID);
  }

  // ---------------- Layer 2 (H=1, C=64) ----------------
  {
    float* h2 = h1;  // reuse [N,64]
    const int rowTiles = N >> 4;
    gat_gemm_wmma_f32<128, 4><<<cdiv(rowTiles, 8), TB, 0, stream>>>(
        out1, W2, h2, N);
    gat_attn_scores<64><<<cdiv((long long)N, TB), TB, 0, stream>>>(
        h2, att_src2, att_dst2, a_s, a_d, N, 1);
    gat_init_stats<<<cdiv((long long)N, TB), TB, 0, stream>>>(mx, den, N);
    float* out2 = (float*)d_out;  // final output [N,64]
    gat_init_bias<<<cdiv((long long)N * OUT, TB), TB, 0, stream>>>(out2, b2, N,
                                                                   OUT);
    gat_edge_logits<1><<<cdiv(ET, TB), TB, 0, stream>>>(src, dst, E, ET, a_s,
                                                        a_d, logits, mx);
    gat_edge_exp<1><<<cdiv(ET, TB), TB, 0, stream>>>(dst, E, ET, logits, mx,
                                                     den);
    gat_edge_agg<1, 64><<<cdiv(ET, TB), TB, 0, stream>>>(src, dst, E, ET, h2,
                                                         logits, den, out2);
  }
}